// GraphConvNet_22935125360678
// MI455X (gfx1250) — compile-verified
//
#include <hip/hip_runtime.h>
#include <hip/hip_bf16.h>

#define N_NODES   50000
#define N_EDGES   1250000
#define N_GRAPHS  64
#define EMB       16
#define HID       64
#define OUT_CH    64
#define DEMO      5
#define MODEL_DIM 16
#define OUT_DIM   2

typedef __attribute__((ext_vector_type(2))) float v2f;
typedef __attribute__((ext_vector_type(8))) float v8f;

// ---------------------------------------------------------------- utilities
__global__ void zero_kernel(float* __restrict__ p, int n) {
    int i = blockIdx.x * blockDim.x + threadIdx.x;
    int stride = gridDim.x * blockDim.x;
    for (; i < n; i += stride) p[i] = 0.0f;
}

// x[node][e] = emb_table[node_ids[node]][e]
__global__ void gather_emb(const int* __restrict__ ids,
                           const float* __restrict__ table,
                           float* __restrict__ x) {
    int i = blockIdx.x * blockDim.x + threadIdx.x;     // over N_NODES*EMB
    if (i >= N_NODES * EMB) return;
    int node = i >> 4;
    int f    = i & 15;
    x[i] = table[ids[node] * EMB + f];
}

// ------------------------------------------------------------- edge scatter
// agg[dst] += w[e] * x[src]; each thread handles 4 contiguous features.
template<int F>
__global__ void scatter_edges(const float* __restrict__ x,
                              const int*   __restrict__ src,
                              const int*   __restrict__ dst,
                              const float* __restrict__ w,
                              float*       __restrict__ agg) {
    constexpr int PE = F / 4;                          // thread-groups per edge
    int idx = blockIdx.x * blockDim.x + threadIdx.x;
    int e  = idx / PE;
    if (e >= N_EDGES) return;
    int f4 = (idx % PE) * 4;
    int s = src[e];
    int d = dst[e];
    float ww = w[e];
    const float4 v = *(const float4*)(x + s * F + f4);
    float* o = agg + d * F + f4;
    unsafeAtomicAdd(o + 0, v.x * ww);
    unsafeAtomicAdd(o + 1, v.y * ww);
    unsafeAtomicAdd(o + 2, v.z * ww);
    unsafeAtomicAdd(o + 3, v.w * ww);
}

// ------------------------------------------------- dense transform via WMMA
// out[m][n] = sum_k agg[m][k]*Wr[n][k] + br[n] + sum_k x[m][k]*Wo[n][k]
// One wave: 16-node (M) x 16-channel (N) tile, f32 WMMA 16x16x4, K step 4.
// Block = 128 threads = 4 waves -> the 4 N-tiles of the 64 output channels.
template<int FIN>
__global__ void __launch_bounds__(128)
graphconv_dense(const float* __restrict__ agg, const float* __restrict__ x,
                const float* __restrict__ Wr,  const float* __restrict__ br,
                const float* __restrict__ Wo,  float* __restrict__ out) {
    const int lane = threadIdx.x & 31;
    const int n0   = (threadIdx.x >> 5) * 16;          // output-channel tile
    const int m0   = blockIdx.x * 16;                  // node tile (exact: 3125*16)
    const int lr   = lane & 15;
    const int kh   = (lane >> 4) * 2;                  // K sub-offset: 0 or 2

    // A 16x4 f32: lane(0-15) VGPR0/1 = K 0/1 for row M=lr ; lanes 16-31 K 2/3.
    // B 4x16 f32 (= W^T): lane holds B[kh+j][n0+lr] = W[n0+lr][kh+j] (contig).
    const float* arow = agg + (m0 + lr) * FIN + kh;
    const float* xrow = x   + (m0 + lr) * FIN + kh;
    const float* wrr  = Wr  + (n0 + lr) * FIN + kh;
    const float* wor  = Wo  + (n0 + lr) * FIN + kh;

    v8f acc = {};
#pragma unroll
    for (int k = 0; k < FIN; k += 4) {
        v2f a, b;
        a.x = arow[k]; a.y = arow[k + 1];
        b.x = wrr[k];  b.y = wrr[k + 1];
        acc = __builtin_amdgcn_wmma_f32_16x16x4_f32(false, a, false, b,
                                                    (short)0, acc, false, false);
        a.x = xrow[k]; a.y = xrow[k + 1];
        b.x = wor[k];  b.y = wor[k + 1];
        acc = __builtin_amdgcn_wmma_f32_16x16x4_f32(false, a, false, b,
                                                    (short)0, acc, false, false);
    }

    // C/D layout: VGPR r -> M = r + 8*(lane/16), N = lane%16
    const float bias  = br[n0 + lr];
    const int   half8 = (lane >> 4) * 8;
#pragma unroll
    for (int r = 0; r < 8; ++r) {
        out[(m0 + half8 + r) * HID + n0 + lr] = acc[r] + bias;
    }
}

// ------------------------------------------------------------- mean pooling
__global__ void pool_sum(const float* __restrict__ x,
                         const int*   __restrict__ batch,
                         float*       __restrict__ gsum) {
    int i = blockIdx.x * blockDim.x + threadIdx.x;     // over N_NODES*HID
    if (i >= N_NODES * HID) return;
    int node = i >> 6;
    int f    = i & 63;
    unsafeAtomicAdd(&gsum[batch[node] * HID + f], x[i]);
}

__global__ void pool_cnt(const int* __restrict__ batch, float* __restrict__ gcnt) {
    int i = blockIdx.x * blockDim.x + threadIdx.x;
    if (i >= N_NODES) return;
    unsafeAtomicAdd(&gcnt[batch[i]], 1.0f);
}

// ----------------------------------------------------------------- MLP head
__global__ void head_mlp(const float* __restrict__ gsum, const float* __restrict__ gcnt,
                         const float* __restrict__ demo,
                         const float* __restrict__ W1, const float* __restrict__ b1,
                         const float* __restrict__ W2, const float* __restrict__ b2,
                         float* __restrict__ out) {
    int g = threadIdx.x;
    if (g >= N_GRAPHS) return;
    float inv = 1.0f / fmaxf(gcnt[g], 1.0f);
    float comb[OUT_CH + DEMO];
#pragma unroll
    for (int f = 0; f < OUT_CH; ++f) comb[f] = gsum[g * OUT_CH + f] * inv;
#pragma unroll
    for (int j = 0; j < DEMO; ++j)   comb[OUT_CH + j] = demo[g * DEMO + j];
    float h[MODEL_DIM];
#pragma unroll
    for (int o = 0; o < MODEL_DIM; ++o) {
        float s = b1[o];
        for (int f = 0; f < OUT_CH + DEMO; ++f)
            s = fmaf(W1[o * (OUT_CH + DEMO) + f], comb[f], s);
        h[o] = fmaxf(s, 0.0f);                         // ReLU (dropout = identity)
    }
#pragma unroll
    for (int o = 0; o < OUT_DIM; ++o) {
        float s = b2[o];
        for (int f = 0; f < MODEL_DIM; ++f)
            s = fmaf(W2[o * MODEL_DIM + f], h[f], s);
        out[g * OUT_DIM + o] = s;
    }
}

// ------------------------------------------------------------------- launch
extern "C" void kernel_launch(void* const* d_in, const int* in_sizes, int n_in,
                              void* d_out, int out_size, void* d_ws, size_t ws_size,
                              hipStream_t stream) {
    const int*   node_ids    = (const int*)  d_in[0];
    const int*   edge_index  = (const int*)  d_in[1];   // [2, E] row-major
    const float* edge_weight = (const float*)d_in[2];
    const int*   batch       = (const int*)  d_in[3];
    const float* demo        = (const float*)d_in[4];
    const float* emb         = (const float*)d_in[5];

    const float *Wr[4], *br[4], *Wo[4], *W1, *b1, *W2, *b2;
    if (n_in >= 22) {   // weight lists pytree-expanded into separate inputs
        for (int i = 0; i < 4; ++i) Wr[i] = (const float*)d_in[6 + i];
        for (int i = 0; i < 4; ++i) br[i] = (const float*)d_in[10 + i];
        for (int i = 0; i < 4; ++i) Wo[i] = (const float*)d_in[14 + i];
        W1 = (const float*)d_in[18]; b1 = (const float*)d_in[19];
        W2 = (const float*)d_in[20]; b2 = (const float*)d_in[21];
    } else {            // weight lists concatenated flat per dict entry
        const float* wrel  = (const float*)d_in[6];
        const float* brel  = (const float*)d_in[7];
        const float* wroot = (const float*)d_in[8];
        const size_t woff[4] = {0, (size_t)HID*EMB, (size_t)HID*EMB + (size_t)HID*HID,
                                (size_t)HID*EMB + 2u*HID*HID};
        for (int i = 0; i < 4; ++i) {
            Wr[i] = wrel  + woff[i];
            Wo[i] = wroot + woff[i];
            br[i] = brel  + (size_t)HID * i;
        }
        W1 = (const float*)d_in[9];  b1 = (const float*)d_in[10];
        W2 = (const float*)d_in[11]; b2 = (const float*)d_in[12];
    }

    const int* src = edge_index;            // edge_index[0, :]
    const int* dst = edge_index + N_EDGES;  // edge_index[1, :]

    float* ws   = (float*)d_ws;
    float* x0   = ws;                                  // N * 16
    float* xa   = x0  + (size_t)N_NODES * EMB;         // N * 64
    float* xb   = xa  + (size_t)N_NODES * HID;         // N * 64
    float* agg  = xb  + (size_t)N_NODES * HID;         // N * 64
    float* gsum = agg + (size_t)N_NODES * HID;         // 64 * 64
    float* gcnt = gsum + (size_t)N_GRAPHS * HID;       // 64  (contiguous w/ gsum)

    // Embedding gather
    gather_emb<<<(N_NODES * EMB + 255) / 256, 256, 0, stream>>>(node_ids, emb, x0);

    // Layer 1: 16 -> 64
    zero_kernel<<<512, 256, 0, stream>>>(agg, N_NODES * EMB);
    scatter_edges<EMB><<<(N_EDGES * (EMB / 4) + 255) / 256, 256, 0, stream>>>(
        x0, src, dst, edge_weight, agg);
    graphconv_dense<EMB><<<N_NODES / 16, 128, 0, stream>>>(
        agg, x0, Wr[0], br[0], Wo[0], xa);

    // Layers 2-4: 64 -> 64 (ping-pong xa/xb)
    float* cur = xa;
    float* nxt = xb;
    for (int L = 1; L < 4; ++L) {
        zero_kernel<<<1024, 256, 0, stream>>>(agg, N_NODES * HID);
        scatter_edges<HID><<<(N_EDGES * (HID / 4) + 255) / 256, 256, 0, stream>>>(
            cur, src, dst, edge_weight, agg);
        graphconv_dense<HID><<<N_NODES / 16, 128, 0, stream>>>(
            agg, cur, Wr[L], br[L], Wo[L], nxt);
        float* t = cur; cur = nxt; nxt = t;
    }

    // Global mean pool (gsum and gcnt are contiguous -> one zero pass)
    zero_kernel<<<16, 256, 0, stream>>>(gsum, N_GRAPHS * HID + N_GRAPHS);
    pool_sum<<<(N_NODES * HID + 255) / 256, 256, 0, stream>>>(cur, batch, gsum);
    pool_cnt<<<(N_NODES + 255) / 256, 256, 0, stream>>>(batch, gcnt);

    // Head MLP -> d_out [64, 2] f32
    head_mlp<<<1, 64, 0, stream>>>(gsum, gcnt, demo, W1, b1, W2, b2, (float*)d_out);
}